// VectorQuantization_7696581394899
// MI455X (gfx1250) — compile-verified
//
#include <hip/hip_runtime.h>
#include <stdint.h>

// Problem constants (from the reference): x [16,4096,256] f32, embedding [1024,256] f32.
#define DDIM   256
#define KCODES 1024
#define NROWS  65536
#define CHUNK  64                     // codes staged per LDS chunk
#define NCHUNK (KCODES / CHUNK)       // 16
#define LDSROW 260                    // 256 + 4 DWORD pad (TDM pad feature) -> conflict-free ds reads
#define ROWS_PER_BLOCK 128            // 8 waves x 16 rows

typedef float v2f  __attribute__((ext_vector_type(2)));
typedef float v4f  __attribute__((ext_vector_type(4)));
typedef float v8f  __attribute__((ext_vector_type(8)));
typedef unsigned int u32x4 __attribute__((ext_vector_type(4)));
typedef int  i32x4 __attribute__((ext_vector_type(4)));
typedef int  i32x8 __attribute__((ext_vector_type(8)));

// Issue a TDM 2D tile load (Global f32 tensor -> LDS), with per-row LDS padding of
// 4 DWORDs every 256 DWORDs so the staged rows have stride LDSROW (bank-conflict-free).
__device__ __forceinline__ void tdm_load_rows_f32(uint32_t lds_addr, const float* gsrc,
                                                  uint32_t tile_rows) {
  uint64_t ga = (uint64_t)(uintptr_t)gsrc;
  u32x4 g0;
  g0[0] = 1u;                                   // count=1, user descriptor, no gather
  g0[1] = lds_addr;                             // lds_addr [63:32]
  g0[2] = (uint32_t)ga;                         // global_addr [95:64]
  g0[3] = (uint32_t)((ga >> 32) & 0x01FFFFFFu)  // global_addr [120:96]
        | (2u << 30);                           // type=2 ("image")
  i32x8 g1;
  g1[0] = (int)((2u << 16)                      // data_size = 4 bytes
        | (1u << 20)                            // pad_enable
        | (7u << 22)                            // pad_interval: 256 DWORDs
        | (3u << 25));                          // pad_amount: 4 DWORDs
  g1[1] = (int)((DDIM & 0xFFFFu) << 16);        // tensor_dim0[15:0]
  g1[2] = (int)((DDIM >> 16) | ((KCODES & 0xFFFFu) << 16));  // dim0 hi | tensor_dim1 lo
  g1[3] = (int)((KCODES >> 16) | ((DDIM & 0xFFFFu) << 16));  // dim1 hi | tile_dim0 = 256
  g1[4] = (int)(tile_rows & 0xFFFFu);           // tile_dim1 = 64 rows (tile_dim2 = 0)
  g1[5] = (int)DDIM;                            // tensor_dim0_stride = 256 elements
  g1[6] = 0;
  g1[7] = 0;
  i32x4 z4 = {0, 0, 0, 0};
#if defined(__clang_major__) && (__clang_major__ >= 23)
  i32x8 z8 = {0, 0, 0, 0, 0, 0, 0, 0};
  __builtin_amdgcn_tensor_load_to_lds(g0, g1, z4, z4, z8, 0);
#else
  __builtin_amdgcn_tensor_load_to_lds(g0, g1, z4, z4, 0);
#endif
}

// Pre-pass: e_sq[k] = sum_d embedding[k][d]^2  (1 MB read; negligible)
extern "C" __global__ void vq_esq_kernel(const float* __restrict__ emb,
                                         float* __restrict__ esq) {
  int k = blockIdx.x * blockDim.x + threadIdx.x;
  if (k < KCODES) {
    const float* e = emb + (size_t)k * DDIM;
    float s = 0.f;
    #pragma unroll 8
    for (int d = 0; d < DDIM; ++d) s += e[d] * e[d];
    esq[k] = s;
  }
}

// Main kernel. ~210 VGPRs/wave by design (x tile register-resident); with a 130 KB
// LDS footprint we run ~1 workgroup per WGP regardless, so relax the occupancy floor
// to let the allocator keep the A fragments in registers instead of spilling.
extern "C" __global__ void
__launch_bounds__(256)
__attribute__((amdgpu_flat_work_group_size(256, 256), amdgpu_waves_per_eu(1)))
vq_main_kernel(const float* __restrict__ x,
               const float* __restrict__ emb,
               const float* __restrict__ esq,
               float* __restrict__ qout,
               float* __restrict__ iout) {
  extern __shared__ float smem[];               // 2 x (64 x LDSROW) f32 = 130 KB
  const int tid  = threadIdx.x;
  const int lane = tid & 31;
  const int ln   = lane & 15;                   // column within 16-lane half
  const int hi   = lane >> 4;                   // which half of the wave
  const int rowBase = blockIdx.x * ROWS_PER_BLOCK + (tid >> 5) * 16;
  const int myRow   = rowBase + ln;

  // Kick off DMA of codebook chunk 0 into LDS buffer 0 (wave 0 drives the TDM).
  if (tid < 32) {
    tdm_load_rows_f32((uint32_t)(uintptr_t)&smem[0], emb, CHUNK);
  }

  // Load this wave's 16x256 x-tile once into registers, pre-swizzled as WMMA
  // f32 16x16x4 A fragments: lane(ln,hi) holds {x[row][4k+2hi], x[row][4k+2hi+1]}.
  v2f a[64];
  {
    const float* xr = x + (size_t)myRow * DDIM + hi * 2;
    #pragma unroll
    for (int kk = 0; kk < 64; ++kk)
      a[kk] = *reinterpret_cast<const v2f*>(xr + kk * 4);
  }

  float best[8];
  int   bidx[8];
  #pragma unroll
  for (int r = 0; r < 8; ++r) { best[r] = -3.0e38f; bidx[r] = 0; }

  #pragma clang loop unroll(disable)
  for (int c = 0; c < NCHUNK; ++c) {
    // Double-buffered TDM: prefetch chunk c+1 while chunk c is consumed.
    if (tid < 32) {
      if (c + 1 < NCHUNK) {
        tdm_load_rows_f32((uint32_t)(uintptr_t)&smem[((c + 1) & 1) * (CHUNK * LDSROW)],
                          emb + (size_t)(c + 1) * CHUNK * DDIM, CHUNK);
        __builtin_amdgcn_s_wait_tensorcnt(1);   // chunk c has landed
      } else {
        __builtin_amdgcn_s_wait_tensorcnt(0);
      }
    }
    __syncthreads();

    const float* buf = &smem[(c & 1) * (CHUNK * LDSROW)];
    // B fragments (4x16, K-major like A): lane(ln,hi) reads code row (t*16+ln).
    const float* b0 = buf + (0 * 16 + ln) * LDSROW + hi * 2;
    const float* b1 = buf + (1 * 16 + ln) * LDSROW + hi * 2;
    const float* b2 = buf + (2 * 16 + ln) * LDSROW + hi * 2;
    const float* b3 = buf + (3 * 16 + ln) * LDSROW + hi * 2;

    v8f acc[4] = {};
    #pragma unroll
    for (int kk = 0; kk < 64; ++kk) {
      v2f av = a[kk];
      acc[0] = __builtin_amdgcn_wmma_f32_16x16x4_f32(
          false, av, false, *reinterpret_cast<const v2f*>(b0 + kk * 4),
          (short)0, acc[0], false, false);
      acc[1] = __builtin_amdgcn_wmma_f32_16x16x4_f32(
          false, av, false, *reinterpret_cast<const v2f*>(b1 + kk * 4),
          (short)0, acc[1], false, false);
      acc[2] = __builtin_amdgcn_wmma_f32_16x16x4_f32(
          false, av, false, *reinterpret_cast<const v2f*>(b2 + kk * 4),
          (short)0, acc[2], false, false);
      acc[3] = __builtin_amdgcn_wmma_f32_16x16x4_f32(
          false, av, false, *reinterpret_cast<const v2f*>(b3 + kk * 4),
          (short)0, acc[3], false, false);
    }

    // Score = 2*dot - ||e||^2 ; ||x||^2 is row-constant and cannot change argmax.
    #pragma unroll
    for (int t = 0; t < 4; ++t) {
      const int code = c * CHUNK + t * 16 + ln;
      const float es = esq[code];
      #pragma unroll
      for (int r = 0; r < 8; ++r) {
        float s = 2.0f * acc[t][r] - es;
        if (s > best[r]) { best[r] = s; bidx[r] = code; }
      }
    }
    __syncthreads();                            // buf[c&1] reusable at iter c+1
  }

  // Cross-lane argmax over the 16 lanes of each half (first-max tie-break),
  // then write index + gather the winning embedding row.
  #pragma unroll
  for (int r = 0; r < 8; ++r) {
    float s = best[r];
    int   idx = bidx[r];
    #pragma unroll
    for (int off = 8; off >= 1; off >>= 1) {
      float s2 = __shfl_xor(s, off, 16);
      int   i2 = __shfl_xor(idx, off, 16);
      if (s2 > s || (s2 == s && i2 < idx)) { s = s2; idx = i2; }
    }
    const int row = rowBase + hi * 8 + r;       // C/D layout: VGPR r <-> M = r + 8*hi
    if (ln == 0) iout[row] = (float)idx;        // d_out is float (f32 outputs, concat)
    const v4f* src = reinterpret_cast<const v4f*>(emb  + (size_t)idx * DDIM);
    v4f*       dst = reinterpret_cast<v4f*>(qout + (size_t)row * DDIM);
    #pragma unroll
    for (int j = 0; j < 4; ++j)
      dst[ln + j * 16] = src[ln + j * 16];      // 16 lanes x 4 float4 = 256 floats
  }
}

extern "C" void kernel_launch(void* const* d_in, const int* in_sizes, int n_in,
                              void* d_out, int out_size, void* d_ws, size_t ws_size,
                              hipStream_t stream) {
  (void)in_sizes; (void)n_in; (void)out_size; (void)ws_size;
  const float* x   = (const float*)d_in[0];
  const float* emb = (const float*)d_in[1];
  float* qout = (float*)d_out;                       // [16,4096,256] f32
  float* iout = qout + (size_t)NROWS * DDIM;         // indices (as float), concat'd
  float* esq  = (float*)d_ws;                        // 1024 f32 scratch

  vq_esq_kernel<<<KCODES / 256, 256, 0, stream>>>(emb, esq);

  const size_t lds_bytes = 2u * CHUNK * LDSROW * sizeof(float); // 133,120 B
  vq_main_kernel<<<NROWS / ROWS_PER_BLOCK, 256, lds_bytes, stream>>>(
      x, emb, esq, qout, iout);
}